// Net_23630910062644
// MI455X (gfx1250) — compile-verified
//
#include <hip/hip_runtime.h>

#define N_FEAT 602
#define HID    128
#define NCLS   41
#define NCLS_P 48
#define KPAD   640   // N_FEAT padded up to multiple of 64

typedef __attribute__((ext_vector_type(2))) float v2f;
typedef __attribute__((ext_vector_type(8))) float v8f;
typedef __attribute__((ext_vector_type(4))) unsigned int u32x4;
typedef __attribute__((ext_vector_type(4))) int i32x4v;
typedef __attribute__((ext_vector_type(8))) int i32x8v;

#define USE_TDM __has_builtin(__builtin_amdgcn_tensor_load_to_lds)

#if USE_TDM
// Issue a TDM 2-D tile load: 16 rows x 64 cols of f32, row stride 602 elements,
// LDS-padded by 4 DWORDs after every 64 DWORDs (-> LDS row stride 68 floats).
// remCols/remRows = tensor extent remaining from the tile origin (OOB reads -> 0).
__device__ __forceinline__ void tdm_load_tile(const float* gsrc, unsigned ldsOff,
                                              int remCols, int remRows) {
    unsigned long long ga = (unsigned long long)gsrc;
    u32x4 g0;
    g0[0] = 1u;                                   // count=1, user descriptor
    g0[1] = ldsOff;                               // lds_addr (bytes)
    g0[2] = (unsigned)ga;                         // global_addr[31:0]
    g0[3] = (unsigned)(ga >> 32) | 0x80000000u;   // global_addr[56:32] | type=2
    i32x8v g1;
    g1[0] = (2 << 16)        // data_size = 4B
          | (1 << 20)        // pad_enable
          | (5 << 22)        // pad_interval: 64 DWORDs
          | (3 << 25);       // pad_amount : 4 DWORDs
    g1[1] = (remCols & 0xFFFF) << 16;                               // tensor_dim0 lo
    g1[2] = ((remCols >> 16) & 0xFFFF) | ((remRows & 0xFFFF) << 16); // dim0 hi | dim1 lo
    g1[3] = ((remRows >> 16) & 0xFFFF) | (64 << 16);                 // dim1 hi | tile_dim0=64
    g1[4] = 16;              // tile_dim1 = 16 (tile_dim2 = 0)
    g1[5] = N_FEAT;          // tensor_dim0_stride lo (48-bit)
    g1[6] = 0;               // stride hi | tensor_dim1_stride lo
    g1[7] = 0;
    i32x4v z4 = {0, 0, 0, 0};
#if __clang_major__ >= 23
    i32x8v z8 = {0, 0, 0, 0, 0, 0, 0, 0};
    __builtin_amdgcn_tensor_load_to_lds(g0, g1, z4, z4, z8, 0);
#else
    __builtin_amdgcn_tensor_load_to_lds(g0, g1, z4, z4, 0);
#endif
}
#endif

// ---------------------------------------------------------------- weight pad
__global__ void pad_weights_k(const float* __restrict__ W1,
                              const float* __restrict__ W2,
                              float* __restrict__ W1p,
                              float* __restrict__ W2p) {
    int i = blockIdx.x * 256 + threadIdx.x;
    if (i < KPAD * HID) W1p[i] = (i < N_FEAT * HID) ? W1[i] : 0.0f;
    if (i < HID * NCLS_P) {
        int r = i / NCLS_P, c = i - r * NCLS_P;
        W2p[i] = (c < NCLS) ? W2[r * NCLS + c] : 0.0f;
    }
}

// ---------------------------------------------------------------- degrees
__global__ void degree_k(const int* __restrict__ src, const int* __restrict__ dst,
                         float* __restrict__ degOut, float* __restrict__ degIn, int nE) {
    int e = blockIdx.x * 256 + threadIdx.x;
    if (e < nE) {
        atomicAdd(&degOut[src[e]], 1.0f);
        atomicAdd(&degIn[dst[e]], 1.0f);
    }
}

__global__ void norm_k(float* __restrict__ ds, float* __restrict__ dd, int nN) {
    int i = blockIdx.x * 256 + threadIdx.x;
    if (i < nN) {
        ds[i] = 1.0f / sqrtf(fmaxf(ds[i], 1.0f));
        dd[i] = 1.0f / sqrtf(fmaxf(dd[i], 1.0f));
    }
}

// ---------------------------------------------------------------- GEMM1: h1 = (x * nsrc) @ W1   [nN x 602] @ [602 x 128]
// block = 256 threads (8 waves). Block handles 16 rows; wave w handles cols [16w,16w+16).
// x tile staged in LDS via TDM (double-buffered, pipelined); norm_src applied to D rows.
__global__ __launch_bounds__(256) void gemm1_k(const float* __restrict__ x,
                                               const float* __restrict__ W1p,
                                               const float* __restrict__ nsrc,
                                               float* __restrict__ h1, int nN) {
    __shared__ __align__(16) float As[2][16 * 68];  // stride 68 -> conflict-free A reads
    const int tid   = threadIdx.x;
    const int lane  = tid & 31;
    const int wave  = tid >> 5;          // 0..7
    const int mBase = blockIdx.x * 16;
    const int nBase = wave * 16;
    const int half  = lane >> 4;         // 0 or 1
    const int l16   = lane & 15;
    const float* xblk = x + (size_t)mBase * N_FEAT;

    v8f c = {};
#if USE_TDM
    const unsigned ldsBase = (unsigned)(unsigned long long)(const void*)&As[0][0];
    const int nCh = KPAD / 64;           // 10
    if (wave == 0)
        tdm_load_tile(xblk, ldsBase, N_FEAT, nN - mBase);
    for (int ch = 0; ch < nCh; ++ch) {
        if (wave == 0) {
            if (ch + 1 < nCh) {
                tdm_load_tile(xblk + (ch + 1) * 64,
                              ldsBase + (unsigned)(((ch + 1) & 1) * (16 * 68 * 4)),
                              N_FEAT - (ch + 1) * 64, nN - mBase);
                __builtin_amdgcn_s_wait_tensorcnt(1);  // chunk ch complete (in-order)
            } else {
                __builtin_amdgcn_s_wait_tensorcnt(0);
            }
        }
        __syncthreads();
        const float* Asb = As[ch & 1];
        const int kBase = ch * 64;
#pragma unroll
        for (int kk = 0; kk < 64; kk += 4) {
            const int kcol = kk + half * 2;
            v2f a, b;
            a.x = Asb[l16 * 68 + kcol];
            a.y = Asb[l16 * 68 + kcol + 1];
            const int kg = kBase + kcol;             // < KPAD, W1p zero-padded
            b.x = W1p[kg * HID + nBase + l16];
            b.y = W1p[(kg + 1) * HID + nBase + l16];
            c = __builtin_amdgcn_wmma_f32_16x16x4_f32(false, a, false, b,
                                                      (short)0, c, false, false);
        }
        __syncthreads();
    }
#else
    // fallback: manual cooperative fill (pure copy; scaling applied at output)
    const int fr = tid >> 4;             // 0..15
    const int fc = (tid & 15) * 4;       // 0..60
    for (int kBase = 0; kBase < KPAD; kBase += 64) {
#pragma unroll
        for (int j = 0; j < 4; ++j) {
            int kg = kBase + fc + j;
            As[0][fr * 68 + fc + j] = (kg < N_FEAT) ? xblk[(size_t)fr * N_FEAT + kg] : 0.0f;
        }
        __syncthreads();
#pragma unroll
        for (int kk = 0; kk < 64; kk += 4) {
            const int kcol = kk + half * 2;
            v2f a, b;
            a.x = As[0][l16 * 68 + kcol];
            a.y = As[0][l16 * 68 + kcol + 1];
            const int kg = kBase + kcol;
            b.x = W1p[kg * HID + nBase + l16];
            b.y = W1p[(kg + 1) * HID + nBase + l16];
            c = __builtin_amdgcn_wmma_f32_16x16x4_f32(false, a, false, b,
                                                      (short)0, c, false, false);
        }
        __syncthreads();
    }
#endif
    const int row0 = mBase + half * 8;
#pragma unroll
    for (int v = 0; v < 8; ++v)
        h1[(size_t)(row0 + v) * HID + nBase + l16] = c[v] * nsrc[row0 + v];
}

// ---------------------------------------------------------------- layer-1 edge scatter-add (L2 resident)
__global__ void agg1_k(const float* __restrict__ h1, const int* __restrict__ src,
                       const int* __restrict__ dst, float* __restrict__ agg, int nE) {
    int i = blockIdx.x * 256 + threadIdx.x;
    if (i < nE * 32) {
        int e = i >> 5;
        int c = (i & 31) << 2;
        int s = src[e], d = dst[e];
        float4 v = *(const float4*)(h1 + (size_t)s * HID + c);
        float* p = agg + (size_t)d * HID + c;
        atomicAdd(p + 0, v.x);
        atomicAdd(p + 1, v.y);
        atomicAdd(p + 2, v.z);
        atomicAdd(p + 3, v.w);
    }
}

// ---------------------------------------------------------------- fused: *ndst + b1 -> relu -> *nsrc (fold layer-2 input scale)
__global__ void act_k(const float* __restrict__ agg, const float* __restrict__ nsrc,
                      const float* __restrict__ ndst, const float* __restrict__ b1,
                      float* __restrict__ hact, int nN) {
    int i = blockIdx.x * 256 + threadIdx.x;
    if (i < nN * HID) {
        int n = i >> 7, f = i & 127;
        float v = agg[i] * ndst[n] + b1[f];
        hact[i] = fmaxf(v, 0.0f) * nsrc[n];
    }
}

// ---------------------------------------------------------------- GEMM2: h2 = hact @ W2p   [nN x 128] @ [128 x 48]
// block = 128 threads (4 waves); each wave owns one 16x16 output tile.
__global__ __launch_bounds__(128) void gemm2_k(const float* __restrict__ hact,
                                               const float* __restrict__ W2p,
                                               float* __restrict__ h2, int nTilesM) {
    const int w = blockIdx.x * 4 + (threadIdx.x >> 5);
    if (w >= nTilesM * 3) return;            // wave-uniform exit, EXEC stays full otherwise
    const int mt = w / 3, nt = w - mt * 3;
    const int lane = threadIdx.x & 31;
    const int half = lane >> 4, l16 = lane & 15;
    const float* arow = hact + (size_t)(mt * 16 + l16) * HID;
    const int col = nt * 16 + l16;

    v8f c = {};
#pragma unroll
    for (int k = 0; k < HID; k += 4) {
        const int kh = k + half * 2;
        v2f a, b;
        a.x = arow[kh];
        a.y = arow[kh + 1];
        b.x = W2p[kh * NCLS_P + col];
        b.y = W2p[(kh + 1) * NCLS_P + col];
        c = __builtin_amdgcn_wmma_f32_16x16x4_f32(false, a, false, b,
                                                  (short)0, c, false, false);
    }
    const int row0 = mt * 16 + half * 8;
#pragma unroll
    for (int v = 0; v < 8; ++v)
        h2[(size_t)(row0 + v) * NCLS_P + col] = c[v];
}

// ---------------------------------------------------------------- layer-2 edge scatter-add
__global__ void agg2_k(const float* __restrict__ h2, const int* __restrict__ src,
                       const int* __restrict__ dst, float* __restrict__ agg, int nE) {
    int i = blockIdx.x * 256 + threadIdx.x;
    if (i < nE * 64) {
        int e = i >> 6, f = i & 63;
        if (f < NCLS)
            atomicAdd(&agg[(size_t)dst[e] * NCLS + f], h2[(size_t)src[e] * NCLS_P + f]);
    }
}

// ---------------------------------------------------------------- final: out = agg * ndst + b2
__global__ void final_k(const float* __restrict__ agg, const float* __restrict__ ndst,
                        const float* __restrict__ b2, float* __restrict__ out, int nN) {
    int i = blockIdx.x * 256 + threadIdx.x;
    if (i < nN * NCLS) {
        int n = i / NCLS, f = i - n * NCLS;
        out[i] = agg[i] * ndst[n] + b2[f];
    }
}

extern "C" void kernel_launch(void* const* d_in, const int* in_sizes, int n_in,
                              void* d_out, int out_size, void* d_ws, size_t ws_size,
                              hipStream_t stream) {
    const float* x    = (const float*)d_in[0];
    const int*   esrc = (const int*)d_in[1];
    const int*   edst = (const int*)d_in[2];
    const float* W1   = (const float*)d_in[3];
    const float* b1   = (const float*)d_in[4];
    const float* W2   = (const float*)d_in[5];
    const float* b2   = (const float*)d_in[6];
    float* out = (float*)d_out;

    const int nE = in_sizes[1];
    const int nN = in_sizes[0] / N_FEAT;     // 100000, multiple of 16

    // workspace carve-up (floats)
    float* ws      = (float*)d_ws;
    float* normSrc = ws;  ws += nN;
    float* normDst = ws;  ws += nN;
    float* W1p     = ws;  ws += (size_t)KPAD * HID;
    float* W2p     = ws;  ws += (size_t)HID * NCLS_P;
    float* bufA    = ws;  ws += (size_t)nN * HID;   // h1, then hact
    float* bufB    = ws;  ws += (size_t)nN * HID;   // agg1, then h2 (stride NCLS_P)
    float* bufC    = ws;                            // agg2 (nN * NCLS)

    // zero the accumulators (graph-capturable async memsets)
    hipMemsetAsync(normSrc, 0, (size_t)2 * nN * sizeof(float), stream);
    hipMemsetAsync(bufB,    0, (size_t)nN * HID * sizeof(float), stream);
    hipMemsetAsync(bufC,    0, (size_t)nN * NCLS * sizeof(float), stream);

    pad_weights_k<<<(KPAD * HID + 255) / 256, 256, 0, stream>>>(W1, W2, W1p, W2p);
    degree_k<<<(nE + 255) / 256, 256, 0, stream>>>(esrc, edst, normSrc, normDst, nE);
    norm_k<<<(nN + 255) / 256, 256, 0, stream>>>(normSrc, normDst, nN);

    gemm1_k<<<nN / 16, 256, 0, stream>>>(x, W1p, normSrc, bufA, nN);
    agg1_k<<<(nE * 32 + 255) / 256, 256, 0, stream>>>(bufA, esrc, edst, bufB, nE);
    act_k<<<(nN * HID + 255) / 256, 256, 0, stream>>>(bufB, normSrc, normDst, b1, bufA, nN);

    const int tiles = (nN / 16) * 3;
    gemm2_k<<<(tiles + 3) / 4, 128, 0, stream>>>(bufA, W2p, bufB, nN / 16);
    agg2_k<<<(nE * 64 + 255) / 256, 256, 0, stream>>>(bufB, esrc, edst, bufC, nE);
    final_k<<<(nN * NCLS + 255) / 256, 256, 0, stream>>>(bufC, normDst, b2, out, nN);
}